// QuantumAnsatz_65481071406602
// MI455X (gfx1250) — compile-verified
//
#include <hip/hip_runtime.h>
#include <math.h>

typedef __attribute__((ext_vector_type(2))) float v2f;
typedef __attribute__((ext_vector_type(8))) float v8f;

#define NW    20
#define HPB   64            // 12-bit high prefixes handled per block
#define WAVES 8
#define ITERS (HPB / WAVES) // 8 WMMA tiles per wave

// amp[b][i] = prod_w (bit_w(i) ? sin : cos) with wire0 = MSB of 20-bit index.
// Factor: i = [hA:6][hB:6][m:4][l:4]; amp = HA[hA]*HB[hB]*R[m]*C[l].
// Each 256-elem tile (fixed h) = outer(Hi*R, C) -> one V_WMMA_F32_16X16X4_F32.
__global__ __launch_bounds__(256)
void qansatz_state_kernel(const float* __restrict__ x,
                          const float* __restrict__ params,
                          float* __restrict__ out)
{
    __shared__ float lc[NW], ls[NW];
    __shared__ float HA[64], HB[64], Rr[16], Cc[16];

    const int tid = threadIdx.x;
    const int bat = blockIdx.y;

    if (tid < NW) {
        float h = (x[bat * NW + tid] + params[tid]) * 0.5f;
        lc[tid] = cosf(h);
        ls[tid] = sinf(h);
    }
    __syncthreads();

    if (tid < 64) {
        float pa = 1.0f, pb = 1.0f;
#pragma unroll
        for (int j = 0; j < 6; ++j) {
            const int bit = (tid >> (5 - j)) & 1;   // wire j <-> MSB-first
            pa *= bit ? ls[j]     : lc[j];
            pb *= bit ? ls[6 + j] : lc[6 + j];
        }
        HA[tid] = pa;
        HB[tid] = pb;
    }
    if (tid < 16) {
        float pr = 1.0f, pc = 1.0f;
#pragma unroll
        for (int j = 0; j < 4; ++j) {
            const int bit = (tid >> (3 - j)) & 1;
            pr *= bit ? ls[12 + j] : lc[12 + j];
            pc *= bit ? ls[16 + j] : lc[16 + j];
        }
        Rr[tid] = pr;
        Cc[tid] = pc;
    }
    __syncthreads();   // past here: no divergence, EXEC all-ones for WMMA

    const int lane = tid & 31;
    const int wave = tid >> 5;
    const int m16  = lane & 15;

    // A-matrix 16x4: lane t holds row M = t&15 (both K slots in its 2 VGPRs).
    const float rA = Rr[m16];
    // B-matrix 4x16: every slot is (some K, N = lane&15). Replicate over all
    // 4 K rows and pre-scale by 1/4 so the dot over K=4 gives exactly R*C.
    const float cB = Cc[m16] * 0.25f;

    const int h0 = blockIdx.x * HPB + wave * ITERS;
    // D layout: VGPR r -> M=r (lanes 0-15) / M=r+8 (lanes 16-31), N = lane&15.
    float* const obat = out + ((size_t)bat << NW)
                            + (((size_t)(lane >> 4)) << 7)   // +128 elems if upper half
                            + (size_t)m16;

#pragma unroll
    for (int it = 0; it < ITERS; ++it) {
        const int   h  = h0 + it;
        const float hi = HA[h >> 6] * HB[h & 63];
        const float av = hi * rA;

        v8f d = {0.f, 0.f, 0.f, 0.f, 0.f, 0.f, 0.f, 0.f};
#if __has_builtin(__builtin_amdgcn_wmma_f32_16x16x4_f32)
        v2f a;  a.x  = av; a.y  = av;    // row M filled across all K
        v2f bm; bm.x = cB; bm.y = cB;    // col N filled across all K, pre-scaled
        d = __builtin_amdgcn_wmma_f32_16x16x4_f32(false, a, false, bm,
                                                  (short)0, d, false, false);
#else
        // VALU fallback producing the identical D register layout.
#pragma unroll
        for (int r = 0; r < 8; ++r) {
            const int M = r + ((lane >> 4) << 3);
            d[r] = 4.0f * (hi * Rr[M]) * cB;   // == hi * R[M] * C[m16]
        }
#endif

        float* o = obat + ((size_t)h << 8);
#pragma unroll
        for (int r = 0; r < 8; ++r)
            o[r * 16] = d[r];   // half-wave-contiguous 64B segments
    }
}

extern "C" void kernel_launch(void* const* d_in, const int* in_sizes, int n_in,
                              void* d_out, int out_size, void* d_ws, size_t ws_size,
                              hipStream_t stream)
{
    const float* x      = (const float*)d_in[0];   // (64, 20) f32
    const float* params = (const float*)d_in[1];   // (20,)    f32
    float*       out    = (float*)d_out;           // 64 * 2^20 f32

    dim3 grid((1 << 12) / HPB, 64);                // 64 h-chunks x 64 batches
    qansatz_state_kernel<<<grid, 256, 0, stream>>>(x, params, out);

    (void)in_sizes; (void)n_in; (void)out_size; (void)d_ws; (void)ws_size;
}